// Bottleneck_5669356834470
// MI455X (gfx1250) — compile-verified
//
#include <hip/hip_runtime.h>
#include <hip/hip_bf16.h>

typedef __attribute__((ext_vector_type(2))) float v2f;
typedef __attribute__((ext_vector_type(8))) float v8f;

#define EPSF 1e-5f
#define HW   3136      // 56*56
#define WDIM 56

// ---------------------------------------------------------------------------
// Kernel A: conv1 (1x1, 256->64) + BN1 + ReLU via V_WMMA_F32_16X16X4_F32
// grid: (196 N-tiles, 16 batch), block: 128 (4 waves = 4 M-tiles of 16)
// ---------------------------------------------------------------------------
__global__ void conv1_bn_relu_wmma(const float* __restrict__ x,
                                   const float* __restrict__ w,     // [64,256]
                                   const float* __restrict__ bg,
                                   const float* __restrict__ bb,
                                   const float* __restrict__ bm,
                                   const float* __restrict__ bv,
                                   float* __restrict__ out1) {
    const int nTile = blockIdx.x;            // 0..195
    const int b     = blockIdx.y;            // 0..15
    const int wave  = threadIdx.x >> 5;      // 0..3 -> M tile
    const int lane  = threadIdx.x & 31;
    const int hi    = lane >> 4;             // 0: K pair {0,1}, 1: K pair {2,3}
    const int l     = lane & 15;
    const int nBase = nTile * 16;
    const int mBase = wave * 16;
    const float* xb = x + (size_t)b * 256 * HW;

    v8f acc = {};
    for (int k = 0; k < 256; k += 4) {
        const int ka = k + (hi ? 2 : 0);
        v2f a, bmat;
        a.x    = w[(mBase + l) * 256 + ka];
        a.y    = w[(mBase + l) * 256 + ka + 1];
        bmat.x = xb[(size_t)ka       * HW + nBase + l];
        bmat.y = xb[(size_t)(ka + 1) * HW + nBase + l];
        acc = __builtin_amdgcn_wmma_f32_16x16x4_f32(false, a, false, bmat,
                                                    (short)0, acc, false, false);
    }
    #pragma unroll
    for (int vv = 0; vv < 8; ++vv) {
        const int mm  = mBase + vv + (hi ? 8 : 0);
        const float s = bg[mm] * rsqrtf(bv[mm] + EPSF);
        float val     = acc[vv] * s + (bb[mm] - bm[mm] * s);
        out1[((size_t)b * 64 + mm) * HW + nBase + l] = fmaxf(val, 0.0f);
    }
}

// ---------------------------------------------------------------------------
// Kernel B: involution weight branch: 1x1(64->16)+BN+ReLU -> 1x1(16->196)+bias
// One thread per pixel. grid: 196 blocks of 256 (= 50176 pixels exactly)
// ---------------------------------------------------------------------------
__global__ void inv_weight_branch(const float* __restrict__ out1,
                                  const float* __restrict__ c1w,   // [16,64]
                                  const float* __restrict__ bg,
                                  const float* __restrict__ bb,
                                  const float* __restrict__ bm,
                                  const float* __restrict__ bv,
                                  const float* __restrict__ c2w,   // [196,16]
                                  const float* __restrict__ c2b,   // [196]
                                  float* __restrict__ wts) {
    const int p  = blockIdx.x * blockDim.x + threadIdx.x;   // 0..50175
    const int b  = p / HW;
    const int sp = p % HW;

    float t[16];
    #pragma unroll
    for (int j = 0; j < 16; ++j) t[j] = 0.0f;

    const float* op = out1 + (size_t)b * 64 * HW + sp;
    for (int c = 0; c < 64; ++c) {
        const float v = op[(size_t)c * HW];
        #pragma unroll
        for (int j = 0; j < 16; ++j) t[j] = fmaf(c1w[j * 64 + c], v, t[j]);
    }
    #pragma unroll
    for (int j = 0; j < 16; ++j) {
        const float s = bg[j] * rsqrtf(bv[j] + EPSF);
        t[j] = fmaxf(t[j] * s + (bb[j] - bm[j] * s), 0.0f);
    }
    float* wp = wts + (size_t)b * 196 * HW + sp;
    for (int o = 0; o < 196; ++o) {
        float s = c2b[o];
        #pragma unroll
        for (int j = 0; j < 16; ++j) s = fmaf(c2w[o * 16 + j], t[j], s);
        wp[(size_t)o * HW] = s;
    }
}

// ---------------------------------------------------------------------------
// Kernel C: involution (K=7, G=4, GC=16) + BN2 + ReLU
// grid: (49 spatial 8x8 tiles, 4 groups, 16 batch), block: 64 threads
// 14x14x16 zero-padded input patch staged in LDS; 49 dyn. weights in regs.
// ---------------------------------------------------------------------------
__global__ void involution_bn_relu(const float* __restrict__ out1,
                                   const float* __restrict__ wts,
                                   const float* __restrict__ bg,
                                   const float* __restrict__ bb,
                                   const float* __restrict__ bm,
                                   const float* __restrict__ bv,
                                   float* __restrict__ out2) {
    __shared__ float patch[16 * 14 * 14];    // 12544 B

    const int tile = blockIdx.x;             // 0..48
    const int g    = blockIdx.y;             // 0..3
    const int b    = blockIdx.z;             // 0..15
    const int ty0  = (tile / 7) * 8;
    const int tx0  = (tile % 7) * 8;
    const int tid  = threadIdx.x;            // 0..63

    const float* src = out1 + ((size_t)b * 64 + g * 16) * HW;
    for (int i = tid; i < 16 * 196; i += 64) {
        const int c  = i / 196;
        const int r  = i % 196;
        const int gy = ty0 - 3 + r / 14;
        const int gx = tx0 - 3 + r % 14;
        float v = 0.0f;
        if (gy >= 0 && gy < WDIM && gx >= 0 && gx < WDIM)
            v = src[(size_t)c * HW + gy * WDIM + gx];
        patch[i] = v;
    }
    __syncthreads();

    const int ty = tid >> 3, tx = tid & 7;
    const int y = ty0 + ty,  x = tx0 + tx;

    float wk[49];
    const float* wp = wts + ((size_t)b * 196 + g * 49) * HW + y * WDIM + x;
    #pragma unroll
    for (int k = 0; k < 49; ++k) wk[k] = wp[(size_t)k * HW];

    for (int c = 0; c < 16; ++c) {
        float acc = 0.0f;
        const float* pc = patch + c * 196;
        #pragma unroll
        for (int kh = 0; kh < 7; ++kh)
            #pragma unroll
            for (int kw = 0; kw < 7; ++kw)
                acc = fmaf(wk[kh * 7 + kw], pc[(ty + kh) * 14 + tx + kw], acc);
        const int mm  = g * 16 + c;
        const float s = bg[mm] * rsqrtf(bv[mm] + EPSF);
        const float v = fmaxf(acc * s + (bb[mm] - bm[mm] * s), 0.0f);
        out2[((size_t)b * 64 + mm) * HW + y * WDIM + x] = v;
    }
}

// ---------------------------------------------------------------------------
// Kernel D: conv3 (1x1, 64->256) + BN3 + residual add + ReLU via WMMA
// grid: (196 N-tiles, 16 batch), block: 512 (16 waves = 16 M-tiles of 16)
// ---------------------------------------------------------------------------
__global__ void conv3_bn_res_relu_wmma(const float* __restrict__ out2,
                                       const float* __restrict__ w,   // [256,64]
                                       const float* __restrict__ bg,
                                       const float* __restrict__ bb,
                                       const float* __restrict__ bm,
                                       const float* __restrict__ bv,
                                       const float* __restrict__ identity,
                                       float* __restrict__ out) {
    const int nTile = blockIdx.x;
    const int b     = blockIdx.y;
    const int wave  = threadIdx.x >> 5;      // 0..15 -> M tile
    const int lane  = threadIdx.x & 31;
    const int hi    = lane >> 4;
    const int l     = lane & 15;
    const int nBase = nTile * 16;
    const int mBase = wave * 16;
    const float* xb = out2 + (size_t)b * 64 * HW;

    v8f acc = {};
    #pragma unroll
    for (int k = 0; k < 64; k += 4) {
        const int ka = k + (hi ? 2 : 0);
        v2f a, bmat;
        a.x    = w[(mBase + l) * 64 + ka];
        a.y    = w[(mBase + l) * 64 + ka + 1];
        bmat.x = xb[(size_t)ka       * HW + nBase + l];
        bmat.y = xb[(size_t)(ka + 1) * HW + nBase + l];
        acc = __builtin_amdgcn_wmma_f32_16x16x4_f32(false, a, false, bmat,
                                                    (short)0, acc, false, false);
    }
    #pragma unroll
    for (int vv = 0; vv < 8; ++vv) {
        const int mm  = mBase + vv + (hi ? 8 : 0);
        const float s = bg[mm] * rsqrtf(bv[mm] + EPSF);
        float val     = acc[vv] * s + (bb[mm] - bm[mm] * s);
        const size_t idx = ((size_t)b * 256 + mm) * HW + nBase + l;
        out[idx] = fmaxf(val + identity[idx], 0.0f);
    }
}

// ---------------------------------------------------------------------------
extern "C" void kernel_launch(void* const* d_in, const int* in_sizes, int n_in,
                              void* d_out, int out_size, void* d_ws, size_t ws_size,
                              hipStream_t stream) {
    (void)in_sizes; (void)n_in; (void)out_size; (void)ws_size;

    const float* x        = (const float*)d_in[0];
    const float* conv1_w  = (const float*)d_in[1];
    const float* bn1_g    = (const float*)d_in[2];
    const float* bn1_b    = (const float*)d_in[3];
    const float* bn1_m    = (const float*)d_in[4];
    const float* bn1_v    = (const float*)d_in[5];
    const float* inv_c1_w = (const float*)d_in[6];
    const float* inv_bn_g = (const float*)d_in[7];
    const float* inv_bn_b = (const float*)d_in[8];
    const float* inv_bn_m = (const float*)d_in[9];
    const float* inv_bn_v = (const float*)d_in[10];
    const float* inv_c2_w = (const float*)d_in[11];
    const float* inv_c2_b = (const float*)d_in[12];
    const float* bn2_g    = (const float*)d_in[13];
    const float* bn2_b    = (const float*)d_in[14];
    const float* bn2_m    = (const float*)d_in[15];
    const float* bn2_v    = (const float*)d_in[16];
    const float* conv3_w  = (const float*)d_in[17];
    const float* bn3_g    = (const float*)d_in[18];
    const float* bn3_b    = (const float*)d_in[19];
    const float* bn3_m    = (const float*)d_in[20];
    const float* bn3_v    = (const float*)d_in[21];
    float* out = (float*)d_out;

    // workspace layout (floats): out1 | wts | out2
    float* ws   = (float*)d_ws;
    float* out1 = ws;                               // 16*64*3136  = 3,211,264
    float* wts  = ws + (size_t)16 * 64 * HW;        // 16*196*3136 = 9,834,496
    float* out2 = wts + (size_t)16 * 196 * HW;      // 16*64*3136

    // A: conv1 + BN1 + ReLU  (WMMA f32)
    conv1_bn_relu_wmma<<<dim3(196, 16), 128, 0, stream>>>(
        x, conv1_w, bn1_g, bn1_b, bn1_m, bn1_v, out1);

    // B: involution weight branch
    inv_weight_branch<<<dim3(196), 256, 0, stream>>>(
        out1, inv_c1_w, inv_bn_g, inv_bn_b, inv_bn_m, inv_bn_v,
        inv_c2_w, inv_c2_b, wts);

    // C: involution + BN2 + ReLU
    involution_bn_relu<<<dim3(49, 4, 16), 64, 0, stream>>>(
        out1, wts, bn2_g, bn2_b, bn2_m, bn2_v, out2);

    // D: conv3 + BN3 + residual + ReLU  (WMMA f32)
    conv3_bn_res_relu_wmma<<<dim3(196, 16), 512, 0, stream>>>(
        out2, conv3_w, bn3_g, bn3_b, bn3_m, bn3_v, x, out);
}